// SE3Transform_16698832847083
// MI455X (gfx1250) — compile-verified
//
#include <hip/hip_runtime.h>
#include <stdint.h>

#ifndef __has_builtin
#define __has_builtin(x) 0
#endif

#define SE3_NB   16   // NUM_BATCH
#define SE3_TPAD 20   // padded floats per 4x4 transform in LDS (conflict-free stride)
#define SE3_MAXT 128  // max transforms staged in LDS

typedef int se3_v4i __attribute__((vector_size(16)));

#if __has_builtin(__builtin_amdgcn_global_load_async_to_lds_b128)
#define SE3_HAVE_ASYNC 1
#else
#define SE3_HAVE_ASYNC 0
#endif

#if __has_builtin(__builtin_amdgcn_s_wait_asynccnt)
#define SE3_WAIT_ASYNC() __builtin_amdgcn_s_wait_asynccnt(0)
#else
#define SE3_WAIT_ASYNC() asm volatile("s_wait_asynccnt 0" ::: "memory")
#endif

// Stage the transform table (nT 4x4 matrices, contiguous in global) into LDS
// with padding: transform t occupies sT[t*SE3_TPAD .. +15].
__device__ __forceinline__ void stage_trans_to_lds(const float* __restrict__ trans,
                                                   float* __restrict__ sT,
                                                   int nT, int tid) {
#if SE3_HAVE_ASYNC
  // CDNA5 async DMA: each lane moves one 16B chunk global->LDS (ASYNCcnt-tracked).
  for (int j = tid; j < nT * 4; j += 256) {
    const int t = j >> 2;       // transform index
    const int c = j & 3;        // 16B chunk within transform
    const float* src = trans + j * 4;
    float* dst = &sT[t * SE3_TPAD + c * 4];   // 80B stride keeps 16B alignment
    __builtin_amdgcn_global_load_async_to_lds_b128(
        (__attribute__((address_space(1))) se3_v4i*)(uintptr_t)src,
        (__attribute__((address_space(3))) se3_v4i*)(uint32_t)(uintptr_t)dst,
        /*imm offset*/0, /*cpol*/0);
  }
  SE3_WAIT_ASYNC();
#else
  for (int j = tid; j < nT * 16; j += 256) {
    sT[(j >> 4) * SE3_TPAD + (j & 15)] = trans[j];
  }
#endif
}

__device__ __forceinline__ void se3_point(const float* __restrict__ T,
                                          float x, float y, float z,
                                          float& ox, float& oy, float& oz) {
  ox = fmaf(T[0], x, fmaf(T[1], y, fmaf(T[2],  z, T[3])));
  oy = fmaf(T[4], x, fmaf(T[5], y, fmaf(T[6],  z, T[7])));
  oz = fmaf(T[8], x, fmaf(T[9], y, fmaf(T[10], z, T[11])));
}

template<int M>
__global__ __launch_bounds__(256)
void se3_partial_dense(const float* __restrict__ trans,
                       const float* __restrict__ xyz,
                       const int* __restrict__ batch,
                       float* __restrict__ out_pos,
                       int* __restrict__ out_batch,
                       int N) {
  __shared__ float sT[SE3_MAXT * SE3_TPAD];
  const int tid = threadIdx.x;
  stage_trans_to_lds(trans, sT, M * SE3_NB, tid);
  __syncthreads();

  const int i = blockIdx.x * 256 + tid;
  if (i >= N) return;

  const float x = xyz[3 * i + 0];
  const float y = xyz[3 * i + 1];
  const float z = xyz[3 * i + 2];
  const int b = batch[i];

#pragma unroll
  for (int m = 0; m < M; ++m) {
    const float* T = &sT[(m * SE3_NB + b) * SE3_TPAD];
    float ox, oy, oz;
    se3_point(T, x, y, z, ox, oy, oz);
    const size_t o = (size_t)m * (size_t)N + (size_t)i;
    out_pos[3 * o + 0] = ox;
    out_pos[3 * o + 1] = oy;
    out_pos[3 * o + 2] = oz;
    out_batch[o] = b + m * SE3_NB;
  }
}

// Runtime-M fallback (same algorithm, no unroll).
__global__ __launch_bounds__(256)
void se3_partial_dense_dyn(const float* __restrict__ trans,
                           const float* __restrict__ xyz,
                           const int* __restrict__ batch,
                           float* __restrict__ out_pos,
                           int* __restrict__ out_batch,
                           int N, int M) {
  __shared__ float sT[SE3_MAXT * SE3_TPAD];
  const int tid = threadIdx.x;
  const int nT = M * SE3_NB;
  stage_trans_to_lds(trans, sT, (nT <= SE3_MAXT) ? nT : SE3_MAXT, tid);
  __syncthreads();

  const int i = blockIdx.x * 256 + tid;
  if (i >= N) return;

  const float x = xyz[3 * i + 0];
  const float y = xyz[3 * i + 1];
  const float z = xyz[3 * i + 2];
  const int b = batch[i];

  for (int m = 0; m < M; ++m) {
    const float* T = &sT[(m * SE3_NB + b) * SE3_TPAD];
    float ox, oy, oz;
    se3_point(T, x, y, z, ox, oy, oz);
    const size_t o = (size_t)m * (size_t)N + (size_t)i;
    out_pos[3 * o + 0] = ox;
    out_pos[3 * o + 1] = oy;
    out_pos[3 * o + 2] = oz;
    out_batch[o] = b + m * SE3_NB;
  }
}

extern "C" void kernel_launch(void* const* d_in, const int* in_sizes, int n_in,
                              void* d_out, int out_size, void* d_ws, size_t ws_size,
                              hipStream_t stream) {
  (void)n_in; (void)out_size; (void)d_ws; (void)ws_size;
  const float* trans = (const float*)d_in[0];   // [M*16, 4, 4] f32
  const float* xyz   = (const float*)d_in[1];   // [N, 3] f32
  const int*   batch = (const int*)d_in[2];     // [N] i32

  const int nT = in_sizes[0] / 16;              // number of 4x4 transforms (128)
  const int M  = nT / SE3_NB;                   // 8
  const int N  = in_sizes[2];                   // 500000

  float* out_pos   = (float*)d_out;                                  // [M*N, 3] f32
  int*   out_batch = (int*)(out_pos + (size_t)3 * (size_t)M * (size_t)N); // [M*N] i32 bits

  dim3 block(256);
  dim3 grid((N + 255) / 256);
  if (M == 8) {
    se3_partial_dense<8><<<grid, block, 0, stream>>>(trans, xyz, batch,
                                                     out_pos, out_batch, N);
  } else {
    se3_partial_dense_dyn<<<grid, block, 0, stream>>>(trans, xyz, batch,
                                                      out_pos, out_batch, N, M);
  }
}